// OrderParameter_29875792511205
// MI455X (gfx1250) — compile-verified
//
#include <hip/hip_runtime.h>
#include <math.h>

typedef __attribute__((ext_vector_type(2))) float v2f;
typedef __attribute__((ext_vector_type(8))) float v8f;

#define KNB 32          // neighbors per atom (fixed by problem / WMMA tiling)
#define WAVES_PER_BLOCK 8

__global__ __launch_bounds__(256) void order_params_kernel(
    const float* __restrict__ vecs,          // [C, 32, 3] f32
    const unsigned char* __restrict__ mask,  // [C, 32] bool (1 byte)
    float* __restrict__ out,                 // [3, C] f32
    int C)
{
  const int lane = threadIdx.x & 31;
  const int wave = threadIdx.x >> 5;
  const int c = blockIdx.x * WAVES_PER_BLOCK + wave;
  if (c >= C) return;                        // wave-uniform guard (EXEC stays all-1)

  const int  l  = lane & 15;
  const bool hi = lane >= 16;

  // ---- validity bitmask (wave-uniform) ----
  const bool mv = mask[(size_t)c * KNB + lane] != 0;
  const unsigned int mbits = (unsigned int)__ballot(mv);

  // ---- load + normalize neighbors l and l+16 ----
  const float* vb = vecs + (size_t)c * KNB * 3;
  float ax = vb[l * 3 + 0],        ay = vb[l * 3 + 1],        az = vb[l * 3 + 2];
  float bx = vb[(l + 16) * 3 + 0], by = vb[(l + 16) * 3 + 1], bz = vb[(l + 16) * 3 + 2];
  float ra = 1.0f / sqrtf(ax * ax + ay * ay + az * az);
  float rb = 1.0f / sqrtf(bx * bx + by * by + bz * bz);
  ax *= ra; ay *= ra; az *= ra;
  bx *= rb; by *= rb; bz *= rb;

  // ---- WMMA A/B tiles (f32 16x4 layout: lanes 0-15 -> K=0,1 ; lanes 16-31 -> K=2,3) ----
  // Gram = V*V^T: identical register content serves as A tile and B tile.
  v2f tlo, thi;
  tlo.x = hi ? az : ax;  tlo.y = hi ? 0.0f : ay;   // neighbors  0..15
  thi.x = hi ? bz : bx;  thi.y = hi ? 0.0f : by;   // neighbors 16..31

  const v8f z8 = {0.f, 0.f, 0.f, 0.f, 0.f, 0.f, 0.f, 0.f};
  v8f d00 = __builtin_amdgcn_wmma_f32_16x16x4_f32(false, tlo, false, tlo, (short)0, z8, false, false);
  v8f d01 = __builtin_amdgcn_wmma_f32_16x16x4_f32(false, tlo, false, thi, (short)0, z8, false, false);
  v8f d10 = __builtin_amdgcn_wmma_f32_16x16x4_f32(false, thi, false, tlo, (short)0, z8, false, false);
  v8f d11 = __builtin_amdgcn_wmma_f32_16x16x4_f32(false, thi, false, thi, (short)0, z8, false, false);

  // ---- constants ----
  const float PI_F    = 3.14159265358979323846f;
  const float THETA0  = 109.47f * PI_F / 180.0f;
  const float INV2DG  = 1.0f / (2.0f * (10.0f * PI_F / 180.0f) * (10.0f * PI_F / 180.0f));
  const float THR     = 160.0f * PI_F / 180.0f;
  const float INV2D1  = 1.0f / (2.0f * (12.0f * PI_F / 180.0f) * (12.0f * PI_F / 180.0f));
  const float INV2D2  = 1.0f / (2.0f * (10.0f * PI_F / 180.0f) * (10.0f * PI_F / 180.0f));

  // ---- element-wise pass over this lane's 32 cos entries ----
  float Stet = 0.f, Sg = 0.f, Soct = 0.f;
  float pa[16], pp[16];                 // per-row: sum(f2*od), sum(lt*od)
#pragma unroll
  for (int i = 0; i < 16; ++i) { pa[i] = 0.f; pp[i] = 0.f; }

  auto proc = [&](float cv, int m, int j, int ridx) {
    float od = ((((mbits >> m) & (mbits >> j)) & 1u) && (m != j)) ? 1.0f : 0.0f;
    cv = fminf(1.0f, fmaxf(-1.0f, cv));
    float t = cv + (1.0f / 3.0f);
    Stet += t * t * od;
    float th = acosf(cv);
    float dg = th - THETA0;
    Sg += __expf(-dg * dg * INV2DG) * od;
    float d1 = th - PI_F;
    float e1 = 3.0f  * __expf(-d1 * d1 * INV2D1);
    float d2 = th - 0.5f * PI_F;
    float e2 = 2.25f * __expf(-d2 * d2 * INV2D2);
    float gt = (th > THR) ? 1.0f : 0.0f;
    float lt = (th < THR) ? 1.0f : 0.0f;
    float f2 = lt * e2;
    Soct += (gt * e1 - f2) * od;        // S1 - Sf2 folded together
    pa[ridx] += f2 * od;
    pp[ridx] += lt * od;
  };

#pragma unroll
  for (int r = 0; r < 8; ++r) {
    int m0 = r + (hi ? 8 : 0);          // rows of D00/D01 tiles
    int m1 = 16 + m0;                   // rows of D10/D11 tiles
    proc(d00[r], m0, l,      r);
    proc(d01[r], m0, l + 16, r);
    proc(d10[r], m1, l,      8 + r);
    proc(d11[r], m1, l + 16, 8 + r);
  }

  // ---- per-row totals: butterfly over the 16-lane half that owns each row ----
#pragma unroll
  for (int ofs = 1; ofs < 16; ofs <<= 1) {
#pragma unroll
    for (int i = 0; i < 16; ++i) {
      pa[i] += __shfl_xor(pa[i], ofs, 16);
      pp[i] += __shfl_xor(pp[i], ofs, 16);
    }
  }
  float dot = 0.f;                      // sum_m row_a[m]*row_lt[m] (16x replicated per half)
#pragma unroll
  for (int i = 0; i < 16; ++i) dot += pa[i] * pp[i];

  // ---- wave-wide scalar reduction ----
#pragma unroll
  for (int ofs = 1; ofs < 32; ofs <<= 1) {
    Stet += __shfl_xor(Stet, ofs, 32);
    Sg   += __shfl_xor(Sg,   ofs, 32);
    Soct += __shfl_xor(Soct, ofs, 32);
    dot  += __shfl_xor(dot,  ofs, 32);
  }
  dot *= (1.0f / 16.0f);                // undo 16x replication within each half

  if (lane == 0) {
    float nf = (float)__popc(mbits);
    float dp = nf * (nf - 1.0f);
    out[c]         = 1.0f - 3.0f * Stet / dp;                       // q_tet
    out[C + c]     = Sg / dp;                                       // q_tetrahedral
    out[2 * C + c] = (Soct + dot) / (nf * (3.0f + (nf - 2.0f) * (nf - 3.0f))); // q_oct
  }
}

extern "C" void kernel_launch(void* const* d_in, const int* in_sizes, int n_in,
                              void* d_out, int out_size, void* d_ws, size_t ws_size,
                              hipStream_t stream) {
  (void)n_in; (void)out_size; (void)d_ws; (void)ws_size;
  const float*         vecs = (const float*)d_in[0];
  const unsigned char* mask = (const unsigned char*)d_in[1];
  float*               out  = (float*)d_out;
  const int C = in_sizes[1] / KNB;      // mask has C*K elements
  const int blocks = (C + WAVES_PER_BLOCK - 1) / WAVES_PER_BLOCK;
  order_params_kernel<<<blocks, 32 * WAVES_PER_BLOCK, 0, stream>>>(vecs, mask, out, C);
}